// Encoder_32942399160406
// MI455X (gfx1250) — compile-verified
//
#include <hip/hip_runtime.h>
#include <hip/hip_bf16.h>

typedef __attribute__((ext_vector_type(2))) float v2f;
typedef __attribute__((ext_vector_type(8))) float v8f;

#define HID 64

// ---------------- degree / normalization precompute ----------------

__global__ void deg_init_kernel(float* __restrict__ deg, int N) {
    int i = blockIdx.x * blockDim.x + threadIdx.x;
    if (i < N) deg[i] = 1.0f;  // self-loop contributes 1
}

__global__ void deg_count_kernel(const int* __restrict__ dst, float* __restrict__ deg, int E) {
    int e = blockIdx.x * blockDim.x + threadIdx.x;
    if (e < E) atomicAdd(&deg[dst[e]], 1.0f);
}

__global__ void deg_rsqrt_kernel(float* __restrict__ deg, int N) {
    int i = blockIdx.x * blockDim.x + threadIdx.x;
    if (i < N) deg[i] = rsqrtf(deg[i]);   // deg buffer becomes dinv
}

__global__ void norm_kernel(const int* __restrict__ src, const int* __restrict__ dst,
                            const float* __restrict__ dinv, float* __restrict__ norm, int E) {
    int e = blockIdx.x * blockDim.x + threadIdx.x;
    if (e < E) norm[e] = dinv[src[e]] * dinv[dst[e]];
}

// ---------------- dense GEMM: Out[M x 64] = act(A)[M x 64|128] @ W[K x 64] ----------------
// One wave computes a full 16x64 row block: 4 accumulators (V_WMMA_F32_16X16X4_F32),
// sharing one A fragment per k-step (2 A loads + 8 B loads per 4 WMMAs).
// A-layout (32-bit A 16x4, ISA 7.12.2): lanes 0-15 hold K=k+0/k+1 for M=lane;
// lanes 16-31 hold K=k+2/k+3. C/D: VGPR j -> row 16*tm + 8*(lane>=16)+j, col (lane&15).

template <bool RELU>
__global__ __launch_bounds__(256) void gemm_wmma_kernel(
        const float* __restrict__ A, const float* __restrict__ W,
        float* __restrict__ Out, int M, int K) {
    const int lane = threadIdx.x & 31;
    const int wib  = threadIdx.x >> 5;
    const int wid  = blockIdx.x * (blockDim.x >> 5) + wib;
    const int n_rowtiles = (M + 15) >> 4;
    if (wid >= n_rowtiles) return;               // wave-uniform: EXEC stays all-ones

    const int hf  = lane >> 4;                   // 0: K pair (0,1); 1: K pair (2,3)
    const int lid = lane & 15;
    const int row = wid * 16 + lid;

    const float* __restrict__ arow = A + (size_t)row * K;
    v8f acc0 = {}, acc1 = {}, acc2 = {}, acc3 = {};

    for (int k = 0; k < K; k += 4) {
        float a0 = arow[k + 2 * hf + 0];
        float a1 = arow[k + 2 * hf + 1];
        if (RELU) { a0 = fmaxf(a0, 0.0f); a1 = fmaxf(a1, 0.0f); }
        v2f a; a.x = a0; a.y = a1;

        const float* __restrict__ wr0 = W + (size_t)(k + 2 * hf + 0) * HID;
        const float* __restrict__ wr1 = W + (size_t)(k + 2 * hf + 1) * HID;
        v2f b0, b1, b2, b3;
        b0.x = wr0[lid +  0]; b0.y = wr1[lid +  0];
        b1.x = wr0[lid + 16]; b1.y = wr1[lid + 16];
        b2.x = wr0[lid + 32]; b2.y = wr1[lid + 32];
        b3.x = wr0[lid + 48]; b3.y = wr1[lid + 48];

        acc0 = __builtin_amdgcn_wmma_f32_16x16x4_f32(false, a, false, b0, (short)0, acc0, false, false);
        acc1 = __builtin_amdgcn_wmma_f32_16x16x4_f32(false, a, false, b1, (short)0, acc1, false, false);
        acc2 = __builtin_amdgcn_wmma_f32_16x16x4_f32(false, a, false, b2, (short)0, acc2, false, false);
        acc3 = __builtin_amdgcn_wmma_f32_16x16x4_f32(false, a, false, b3, (short)0, acc3, false, false);
    }

    float* __restrict__ outp = Out + (size_t)(wid * 16 + hf * 8) * HID + lid;
#pragma unroll
    for (int j = 0; j < 8; ++j) {
        outp[(size_t)j * HID +  0] = acc0[j];
        outp[(size_t)j * HID + 16] = acc1[j];
        outp[(size_t)j * HID + 32] = acc2[j];
        outp[(size_t)j * HID + 48] = acc3[j];
    }
}

// ---------------- self-loop init: agg[i][c] = h2[i][c] * dinv[i]^2 + b[c] (float4) ----------------

__global__ void init_self_kernel(const float* __restrict__ h2, const float* __restrict__ dinv,
                                 const float* __restrict__ bias, float* __restrict__ agg, int N) {
    long long t = (long long)blockIdx.x * blockDim.x + threadIdx.x;   // one float4 per thread
    if (t >= (long long)N * (HID / 4)) return;
    int i  = (int)(t >> 4);              // node
    int c4 = (int)(t & 15) * 4;          // column group
    float w = dinv[i];
    w = w * w;
    const float4 h = *(const float4*)(h2 + (size_t)i * HID + c4);
    const float4 b = *(const float4*)(bias + c4);
    float4 r;
    r.x = h.x * w + b.x;
    r.y = h.y * w + b.y;
    r.z = h.z * w + b.z;
    r.w = h.w * w + b.w;
    *(float4*)(agg + (size_t)i * HID + c4) = r;
}

// ---------------- edge scatter: agg[dst] += h2[src] * norm ----------------
// One wave per edge: edge id is scalar (readfirstlane) so src/dst/norm become
// scalar loads; each lane handles 2 columns via float2 load + 2 f32 atomics.

__global__ __launch_bounds__(256) void edge_agg_kernel(
        const float* __restrict__ h2, const int* __restrict__ src,
        const int* __restrict__ dst, const float* __restrict__ norm,
        float* __restrict__ agg, int E) {
    const int lane = threadIdx.x & 31;
    const int wib  = __builtin_amdgcn_readfirstlane(threadIdx.x >> 5);
    const int e    = blockIdx.x * (blockDim.x >> 5) + wib;
    if (e >= E) return;
    const int   s = src[e];
    const int   d = dst[e];
    const float w = norm[e];
    const float2 v = *(const float2*)(h2 + (size_t)s * HID + 2 * lane);
    float* __restrict__ o = agg + (size_t)d * HID + 2 * lane;
    atomicAdd(o + 0, v.x * w);
    atomicAdd(o + 1, v.y * w);
}

// ---------------- host launch ----------------

extern "C" void kernel_launch(void* const* d_in, const int* in_sizes, int n_in,
                              void* d_out, int out_size, void* d_ws, size_t ws_size,
                              hipStream_t stream) {
    const float* x  = (const float*)d_in[0];
    const int*   ei = (const int*)  d_in[1];
    const float* W1 = (const float*)d_in[2];
    const float* b1 = (const float*)d_in[3];
    const float* W2 = (const float*)d_in[4];
    const float* b2 = (const float*)d_in[5];
    const float* W3 = (const float*)d_in[6];
    const float* b3 = (const float*)d_in[7];
    const float* W4 = (const float*)d_in[8];
    const float* b4 = (const float*)d_in[9];
    float* out = (float*)d_out;

    const int K1 = in_sizes[2] / HID;       // 128
    const int N  = in_sizes[0] / K1;        // 50000
    const int E  = in_sizes[1] / 2;         // 1600000
    const int* src = ei;
    const int* dst = ei + E;

    // workspace layout (f32)
    float* h2   = (float*)d_ws;             // N*64
    float* agg  = h2  + (size_t)N * HID;    // N*64
    float* dinv = agg + (size_t)N * HID;    // N   (deg -> dinv in place)
    float* nrm  = dinv + N;                 // E

    const int B = 256;
    const int wpb = B / 32;                                   // 8 waves per block
    const int gN  = (N + B - 1) / B;
    const int gE  = (E + B - 1) / B;
    const int gI  = (int)(((long long)N * (HID / 4) + B - 1) / B);
    const int gEd = (E + wpb - 1) / wpb;                      // 1 wave per edge
    const int n_rowtiles = (N + 15) / 16;
    const int gG  = (n_rowtiles + wpb - 1) / wpb;

    // normalization precompute
    deg_init_kernel <<<gN, B, 0, stream>>>(dinv, N);
    deg_count_kernel<<<gE, B, 0, stream>>>(dst, dinv, E);
    deg_rsqrt_kernel<<<gN, B, 0, stream>>>(dinv, N);
    norm_kernel     <<<gE, B, 0, stream>>>(src, dst, dinv, nrm, E);

    // layer 1: h2 = x @ W1 ; agg = self + edges
    gemm_wmma_kernel<false><<<gG, B, 0, stream>>>(x, W1, h2, N, K1);
    init_self_kernel<<<gI, B, 0, stream>>>(h2, dinv, b1, agg, N);
    edge_agg_kernel <<<gEd, B, 0, stream>>>(h2, src, dst, nrm, agg, E);

    // layer 2: h2 = relu(agg) @ W2 ; agg = self + edges
    gemm_wmma_kernel<true><<<gG, B, 0, stream>>>(agg, W2, h2, N, HID);
    init_self_kernel<<<gI, B, 0, stream>>>(h2, dinv, b2, agg, N);
    edge_agg_kernel <<<gEd, B, 0, stream>>>(h2, src, dst, nrm, agg, E);

    // layer 3
    gemm_wmma_kernel<true><<<gG, B, 0, stream>>>(agg, W3, h2, N, HID);
    init_self_kernel<<<gI, B, 0, stream>>>(h2, dinv, b3, agg, N);
    edge_agg_kernel <<<gEd, B, 0, stream>>>(h2, src, dst, nrm, agg, E);

    // layer 4 -> d_out (no relu on output)
    gemm_wmma_kernel<true><<<gG, B, 0, stream>>>(agg, W4, h2, N, HID);
    init_self_kernel<<<gI, B, 0, stream>>>(h2, dinv, b4, out, N);
    edge_agg_kernel <<<gEd, B, 0, stream>>>(h2, src, dst, nrm, out, E);
}